// SurfaceDistanceLoss_39565238730835
// MI455X (gfx1250) — compile-verified
//
#include <hip/hip_runtime.h>
#include <math.h>

typedef __attribute__((ext_vector_type(2))) float v2f;
typedef __attribute__((ext_vector_type(8))) float v8f;

#define B_ 64
#define J_ 42
#define P_ 16384
#define JT 3          // j tiles of 16 rows (covers 48 >= 42)
#define SPLITS 8      // point-chunks per batch -> 512 blocks
#define WAVES 8       // waves per block (256 threads, wave32)

// Monotonic float->uint encoding: preserves < ordering incl. negatives.
__device__ __forceinline__ unsigned enc_f32(float f) {
    unsigned u = __float_as_uint(f);
    return (u & 0x80000000u) ? ~u : (u | 0x80000000u);
}
__device__ __forceinline__ float dec_f32(unsigned e) {
    unsigned u = (e & 0x80000000u) ? (e ^ 0x80000000u) : ~e;
    return __uint_as_float(u);
}

// min with DPP-shuffled partner: one v_mov_dpp + one v_min, no LDS.
template <int CTRL>
__device__ __forceinline__ float min_dpp(float v) {
    int x = __builtin_amdgcn_update_dpp(0, __float_as_int(v), CTRL,
                                        0xf, 0xf, true);
    return fminf(v, __int_as_float(x));
}
// Full min over each 16-lane half (assoc. min -> pairing order irrelevant):
// quad xor1, quad xor2, half-row mirror (crosses quads), row mirror (crosses 8s).
__device__ __forceinline__ float min_row16(float v) {
    v = min_dpp<0xB1>(v);    // quad_perm [1,0,3,2]
    v = min_dpp<0x4E>(v);    // quad_perm [2,3,0,1]
    v = min_dpp<0x141>(v);   // row_half_mirror
    v = min_dpp<0x140>(v);   // row_mirror
    return v;
}

__global__ void init_min_kernel(unsigned* wsmin) {
    int i = blockIdx.x * blockDim.x + threadIdx.x;
    if (i < B_ * 48) wsmin[i] = 0xFFFFFFFFu;
}

__global__ __launch_bounds__(256) void cdist_min_kernel(
    const float* __restrict__ pred, const float* __restrict__ pts,
    unsigned* __restrict__ wsmin)
{
    const int b    = blockIdx.x / SPLITS;
    const int s    = blockIdx.x % SPLITS;
    const int wave = threadIdx.x >> 5;
    const int lane = threadIdx.x & 31;
    const int n    = lane & 15;          // row (A) / column (B) index
    const bool hi  = lane >= 16;         // upper half holds K=2,3

    // A fragments: row j holds (-2x, -2y, -2z, 1). Rows >= J_ are (0,0,0,1)
    // (their D values are >= 0 junk; finalizer only reads j < 42).
    v2f afrag[JT];
    #pragma unroll
    for (int t = 0; t < JT; ++t) {
        int j = t * 16 + n;
        float x = 0.f, y = 0.f, z = 0.f;
        if (j < J_) {
            const float* jp = pred + ((size_t)b * J_ + j) * 3;
            x = jp[0]; y = jp[1]; z = jp[2];
        }
        afrag[t][0] = hi ? (-2.f * z) : (-2.f * x);   // K=2 : K=0
        afrag[t][1] = hi ? 1.f        : (-2.f * y);   // K=3 : K=1
    }

    float acc[JT][8];
    #pragma unroll
    for (int t = 0; t < JT; ++t)
        #pragma unroll
        for (int r = 0; r < 8; ++r) acc[t][r] = 3.4e38f;

    const int tilesPerBlock = P_ / 16 / SPLITS;       // 128
    const int tilesPerWave  = tilesPerBlock / WAVES;  // 16
    const int tile0 = s * tilesPerBlock + wave * tilesPerWave;

    #pragma unroll 4
    for (int ti = 0; ti < tilesPerWave; ++ti) {
        int p = (tile0 + ti) * 16 + n;
        const float* pp = pts + ((size_t)b * P_ + p) * 3;
        float x = pp[0], y = pp[1], z = pp[2];
        float b2 = x * x + y * y + z * z;
        v2f bfrag;
        bfrag[0] = hi ? z  : x;      // K=2 : K=0   (col = n)
        bfrag[1] = hi ? b2 : y;      // K=3 : K=1

        #pragma unroll
        for (int t = 0; t < JT; ++t) {
            v8f c = {};
            // D[j,p] = -2*dot(joint_j, point_p) + |point_p|^2
            v8f d = __builtin_amdgcn_wmma_f32_16x16x4_f32(
                false, afrag[t], false, bfrag, (short)0, c, false, false);
            #pragma unroll
            for (int r = 0; r < 8; ++r)
                acc[t][r] = fminf(acc[t][r], d[r]);
        }
    }

    // D layout: VGPR r = row r (lanes 0-15) / row r+8 (lanes 16-31), col = n.
    // DPP min across the 16 columns inside each half, then combine globally.
    #pragma unroll
    for (int t = 0; t < JT; ++t) {
        #pragma unroll
        for (int r = 0; r < 8; ++r) {
            float v = min_row16(acc[t][r]);
            if (n == 0) {   // lanes 0 (row r) and 16 (row r+8)
                int row = t * 16 + r + (hi ? 8 : 0);
                atomicMin(&wsmin[b * 48 + row], enc_f32(v));
            }
        }
    }
}

__global__ __launch_bounds__(256) void finalize_kernel(
    const unsigned* __restrict__ wsmin, const float* __restrict__ pred,
    const float* __restrict__ gt, float* __restrict__ out)
{
    __shared__ float red[256];
    float sum = 0.f;
    for (int i = threadIdx.x; i < B_ * J_; i += 256) {
        int b = i / J_, j = i % J_;
        float minD = dec_f32(wsmin[b * 48 + j]);
        const float* jp = pred + ((size_t)b * J_ + j) * 3;
        float a2 = jp[0]*jp[0] + jp[1]*jp[1] + jp[2]*jp[2];
        float sq = fmaxf(minD + a2, 0.f);   // min_p(a2 + b2 - 2ab), clamped
        float d  = sqrtf(sq);
        float diff = d - gt[i];
        sum += diff * diff;
    }
    red[threadIdx.x] = sum;
    __syncthreads();
    for (int st = 128; st > 0; st >>= 1) {
        if ((int)threadIdx.x < st) red[threadIdx.x] += red[threadIdx.x + st];
        __syncthreads();
    }
    if (threadIdx.x == 0) out[0] = red[0] / (float)(B_ * J_);
}

extern "C" void kernel_launch(void* const* d_in, const int* in_sizes, int n_in,
                              void* d_out, int out_size, void* d_ws, size_t ws_size,
                              hipStream_t stream) {
    const float* pred = (const float*)d_in[0];   // [64,42,3]
    const float* pts  = (const float*)d_in[1];   // [64,16384,3]
    const float* gt   = (const float*)d_in[2];   // [64,42]
    unsigned* wsmin   = (unsigned*)d_ws;         // [64*48] encoded mins (12 KB)

    init_min_kernel<<<(B_ * 48 + 255) / 256, 256, 0, stream>>>(wsmin);
    cdist_min_kernel<<<B_ * SPLITS, 256, 0, stream>>>(pred, pts, wsmin);
    finalize_kernel<<<1, 256, 0, stream>>>(wsmin, pred, gt, (float*)d_out);
}